// Decoder_40793599377426
// MI455X (gfx1250) — compile-verified
//
#include <hip/hip_runtime.h>

// ---- problem dims ----
#define B_      128
#define P_      196
#define ENC_    2048
#define DEC_    512
#define ATT_    512
#define EMB_    512
#define V_      10000
#define MAXLEN_ 26
#define T_      25

typedef __attribute__((ext_vector_type(2))) float v2f;
typedef __attribute__((ext_vector_type(8))) float v8f;

__device__ __forceinline__ float sigmoidf_(float x) { return 1.0f / (1.0f + __expf(-x)); }

// ---------------------------------------------------------------------------
// fp32 WMMA GEMM with LDS-staged B:  C[M,N] (= or +=) A[M,K] @ Bw[K,N] (+bias)
//  - block = 8 waves = 128 M rows x 64 N cols; all waves share one B strip
//  - B chunk (32K x 64N) staged coalesced into LDS, transposed (stride 34 pad)
//    -> fragment reads are 8B-aligned ds_load_b64, <=2-way bank conflicts
//  - K-chunk loop unrolled (8 x 4 WMMAs) for latency hiding
//  - N is always a multiple of 16 here -> per-16-group validity is uniform;
//    out-of-range B cols are loaded at a clamped address (never stored)
//  - optional A row remap (fused argsort gather), accumulate, bias,
//    per-row length mask with strided store (fc head writes (B,T,V) directly)
// ---------------------------------------------------------------------------
#define KB_ 32
#define LBS 34              // padded LDS stride per column (even -> b64 aligned)

__global__ __launch_bounds__(256)
void k_gemm_wmma(const float* __restrict__ A, int lda,
                 const float* __restrict__ Bw, int ldb,
                 const float* __restrict__ bias,
                 float* __restrict__ C, int ldc,
                 int M, int N, int K,
                 const int* __restrict__ rowmap, int rpg,
                 int accum,
                 const int* __restrict__ dec_len, int step_t)
{
    __shared__ float lb[64 * LBS];                 // 8.5 KB B tile

    const int tid  = threadIdx.x;
    const int lane = tid & 31;
    const int wid  = tid >> 5;
    const int n0   = blockIdx.x << 6;              // N strip shared by the block
    const int mt   = blockIdx.y * 8 + wid;         // wave's M tile
    const int m0   = mt << 4;
    const bool wvalid = (m0 < M);                  // wave-uniform

    // A fragment row (constant over K). 16x4 fp32 A layout:
    // lanes 0-15 -> M rows with K+0/K+1, lanes 16-31 same rows with K+2/K+3.
    int arow = m0 + (lane & 15);
    if (arow >= M) arow = M - 1;                   // clamped; store is guarded
    int srow = arow;
    if (rowmap) srow = rowmap[arow / rpg] * rpg + (arow % rpg);
    const float* Ap = A + (long long)srow * lda + ((lane >> 4) << 1);

    const int khw = (lane >> 4) << 1;              // 0 or 2: lane-half K offset
    const int nl  = lane & 15;

    v8f acc[4] = {v8f{0,0,0,0,0,0,0,0}, v8f{0,0,0,0,0,0,0,0},
                  v8f{0,0,0,0,0,0,0,0}, v8f{0,0,0,0,0,0,0,0}};

    for (int k0 = 0; k0 < K; k0 += KB_) {
        __syncthreads();                           // previous chunk fully consumed
        // ---- stage B[k0..k0+31][n0..n0+63] -> lb[col][k], coalesced ----
#pragma unroll
        for (int i = 0; i < (KB_ * 64) / 256; ++i) {
            const int e  = tid + i * 256;
            const int kk = e >> 6;
            const int cc = e & 63;
            int col = n0 + cc;
            if (col >= N) col = N - 1;             // clamp: safe, never stored
            lb[cc * LBS + kk] = Bw[(long long)(k0 + kk) * ldb + col];
        }
        __syncthreads();

        if (wvalid) {
            if (k0 + KB_ < K)                      // hint next A chunk (L2)
                __builtin_prefetch(Ap + k0 + KB_, 0, 1);
#pragma unroll
            for (int ks = 0; ks < KB_; ks += 4) {
                v2f a;
                a.x = Ap[k0 + ks];                 // global_load_b64
                a.y = Ap[k0 + ks + 1];
                const int kk = ks + khw;
#pragma unroll
                for (int j = 0; j < 4; ++j) {
                    const v2f b = *(const v2f*)(lb + (j * 16 + nl) * LBS + kk);
                    acc[j] = __builtin_amdgcn_wmma_f32_16x16x4_f32(
                                 false, a, false, b, (short)0, acc[j], false, false);
                }
            }
        }
    }

    if (!wvalid) return;

    // C/D 16x16 fp32 layout: vgpr r -> M = r (+8 for lanes 16-31), N = lane&15
#pragma unroll
    for (int r = 0; r < 8; ++r) {
        const int m = m0 + r + ((lane >> 4) << 3);
        bool act = true;
        if (dec_len) act = (step_t < dec_len[m]);
#pragma unroll
        for (int j = 0; j < 4; ++j) {
            if ((n0 + (j << 4)) >= N) continue;    // uniform per group (N%16==0)
            const int col = n0 + (j << 4) + nl;
            float v = acc[j][r];
            if (bias) v += bias[col];
            const long long ci = (long long)m * ldc + col;
            if (accum) v += C[ci];
            C[ci] = act ? v : 0.0f;
        }
    }
}

// ---------------------------------------------------------------------------
// stable rank sort matching jnp.argsort(-lengths): descending length, stable
// ---------------------------------------------------------------------------
__global__ void k_sort(const int* __restrict__ cl, int* __restrict__ sind,
                       int* __restrict__ dlen)
{
    const int b = threadIdx.x;
    if (b >= B_) return;
    const int Lb = cl[b];
    int rank = 0;
    for (int j = 0; j < B_; ++j) {
        const int Lj = cl[j];
        if (Lj > Lb || (Lj == Lb && j < b)) rank++;
    }
    sind[rank] = b;
    dlen[rank] = Lb - 1;
}

__global__ void k_addvec(const float* __restrict__ a, const float* __restrict__ b,
                         float* __restrict__ o, int n)
{
    const int i = blockIdx.x * 256 + threadIdx.x;
    if (i < n) o[i] = a[i] + b[i];
}

__global__ void k_zero(float* __restrict__ p, int n)
{
    const int i = blockIdx.x * 256 + threadIdx.x;
    if (i < n) p[i] = 0.0f;
}

// embedding gather in sorted batch order: es[b,t,e] = emb[caps[sind[b],t], e]
__global__ void k_embed(const int* __restrict__ caps, const int* __restrict__ sind,
                        const float* __restrict__ emb, float* __restrict__ es)
{
    const int idx = blockIdx.x * 256 + threadIdx.x;
    if (idx >= B_ * T_ * EMB_) return;
    const int e = idx % EMB_;
    const int t = (idx / EMB_) % T_;
    const int b = idx / (EMB_ * T_);
    const int tok = caps[sind[b] * MAXLEN_ + t];
    es[idx] = emb[(long long)tok * EMB_ + e];
}

// scores[b,p] = dot(relu(att1[b,p,:] + att23[b,:]), full_w) + full_b  (one wave/row)
__global__ __launch_bounds__(256)
void k_scores(const float* __restrict__ att1, const float* __restrict__ att23,
              const float* __restrict__ full_w, const float* __restrict__ full_b,
              float* __restrict__ scores, int rows)
{
    const int lane = threadIdx.x & 31;
    const int row  = blockIdx.x * 8 + (threadIdx.x >> 5);
    if (row >= rows) return;
    const int b = row / P_;
    const float* e1 = att1  + (long long)row * ATT_;
    const float* e2 = att23 + (long long)b * ATT_;
    float s = 0.0f;
    for (int j = lane; j < ATT_; j += 32) {
        float e = e1[j] + e2[j];
        e = fmaxf(e, 0.0f);
        s += e * full_w[j];
    }
#pragma unroll
    for (int off = 16; off > 0; off >>= 1) s += __shfl_xor(s, off, 32);
    if (lane == 0) scores[row] = s + full_b[0];
}

// in-place softmax over P per batch row (one block per b)
__global__ __launch_bounds__(256)
void k_softmax(float* __restrict__ sc)
{
    __shared__ float red[256];
    const int b = blockIdx.x;
    const int t = threadIdx.x;
    const float v = (t < P_) ? sc[b * P_ + t] : -3.4e38f;
    red[t] = v;
    __syncthreads();
    for (int s = 128; s > 0; s >>= 1) {
        if (t < s) red[t] = fmaxf(red[t], red[t + s]);
        __syncthreads();
    }
    const float mx = red[0];
    __syncthreads();
    const float ev = (t < P_) ? __expf(v - mx) : 0.0f;
    red[t] = ev;
    __syncthreads();
    for (int s = 128; s > 0; s >>= 1) {
        if (t < s) red[t] += red[t + s];
        __syncthreads();
    }
    const float inv = 1.0f / red[0];
    if (t < P_) sc[b * P_ + t] = ev * inv;
}

// awe[b,e] = sum_p alpha[b,p] * enc[sind[b],p,e]   (grid: (ENC/256, B))
__global__ __launch_bounds__(256)
void k_awe(const float* __restrict__ alpha, const float* __restrict__ enc,
           const int* __restrict__ sind, float* __restrict__ awe)
{
    __shared__ float sa[P_];
    const int b = blockIdx.y;
    if (threadIdx.x < P_) sa[threadIdx.x] = alpha[b * P_ + threadIdx.x];
    __syncthreads();
    const int e = blockIdx.x * 256 + threadIdx.x;
    const float* ep = enc + (long long)sind[b] * P_ * ENC_ + e;
    float acc = 0.0f;
#pragma unroll 4
    for (int p = 0; p < P_; ++p) acc += sa[p] * ep[(long long)p * ENC_];
    awe[b * ENC_ + e] = acc;
}

// LSTM pointwise: c = sig(f)*c + sig(i)*tanh(g); h = sig(o)*tanh(c); masked carry
__global__ __launch_bounds__(256)
void k_lstm(const float* __restrict__ g, float* __restrict__ h, float* __restrict__ c,
            const int* __restrict__ dec_len, int step_t, int use_cin)
{
    const int idx = blockIdx.x * 256 + threadIdx.x;
    if (idx >= B_ * DEC_) return;
    const int b = idx / DEC_;
    const int j = idx % DEC_;
    if (dec_len && step_t >= dec_len[b]) return;   // keep old carry (jnp.where)
    const float* gr = g + (long long)b * (4 * DEC_);
    const float gi = gr[j];
    const float gf = gr[DEC_ + j];
    const float gg = gr[2 * DEC_ + j];
    const float go = gr[3 * DEC_ + j];
    const float cold = use_cin ? c[idx] : 0.0f;
    const float c2 = sigmoidf_(gf) * cold + sigmoidf_(gi) * tanhf(gg);
    c[idx] = c2;
    h[idx] = sigmoidf_(go) * tanhf(c2);
}

// ---------------------------------------------------------------------------
extern "C" void kernel_launch(void* const* d_in, const int* in_sizes, int n_in,
                              void* d_out, int out_size, void* d_ws, size_t ws_size,
                              hipStream_t stream)
{
    // ---- inputs (setup_inputs() dict order, params in insertion order) ----
    const float* attrs   = (const float*)d_in[0];   // (B, 512)
    const float* enc     = (const float*)d_in[1];   // (B, 196, 2048)
    const int*   caps    = (const int*)  d_in[2];   // (B, 26)
    const int*   cl      = (const int*)  d_in[3];   // (B, 1)
    const float* emb     = (const float*)d_in[4];   // (V, 512)
    const float* x0_w    = (const float*)d_in[5];
    const float* x0_b    = (const float*)d_in[6];
    const float* l1_wih  = (const float*)d_in[7];   // (512, 2048)
    const float* l1_bih  = (const float*)d_in[8];
    const float* l1_whh  = (const float*)d_in[9];   // (512, 2048)
    const float* l1_bhh  = (const float*)d_in[10];
    const float* l2_wih  = (const float*)d_in[11];  // (2560, 2048)
    const float* l2_bih  = (const float*)d_in[12];
    const float* l2_whh  = (const float*)d_in[13];  // (512, 2048)
    const float* l2_bhh  = (const float*)d_in[14];
    const float* ae_w    = (const float*)d_in[15];  // (2048, 512)
    const float* ae_b    = (const float*)d_in[16];
    const float* ad_w    = (const float*)d_in[17];  // (512, 512)
    const float* ad_b    = (const float*)d_in[18];
    const float* aa_w    = (const float*)d_in[19];  // (512, 512)
    const float* aa_b    = (const float*)d_in[20];
    const float* full_w  = (const float*)d_in[21];  // (512,)
    const float* full_b  = (const float*)d_in[22];  // scalar
    const float* fc_w    = (const float*)d_in[23];  // (512, 10000)
    const float* fc_b    = (const float*)d_in[24];  // (10000,)
    float* out = (float*)d_out;                     // (B, T, V)

    // ---- workspace carve (~63 MB) ----
    char* w = (char*)d_ws;
    int*   sind    = (int*)w;                     w += 128 * sizeof(int);
    int*   dlen    = (int*)w;                     w += 128 * sizeof(int);
    float* bias_l1 = (float*)w;                   w += 2048 * sizeof(float);
    float* bias_l2 = (float*)w;                   w += 2048 * sizeof(float);
    float* bias_at = (float*)w;                   w += 512 * sizeof(float);
    float* x0      = (float*)w;                   w += B_ * DEC_ * sizeof(float);
    float* h1      = (float*)w;                   w += B_ * DEC_ * sizeof(float);
    float* c1      = (float*)w;                   w += B_ * DEC_ * sizeof(float);
    float* h2      = (float*)w;                   w += B_ * DEC_ * sizeof(float);
    float* c2      = (float*)w;                   w += B_ * DEC_ * sizeof(float);
    float* att23   = (float*)w;                   w += B_ * ATT_ * sizeof(float);
    float* awe     = (float*)w;                   w += B_ * ENC_ * sizeof(float);
    float* g1      = (float*)w;                   w += B_ * 4 * DEC_ * sizeof(float);
    float* g2      = (float*)w;                   w += B_ * 4 * DEC_ * sizeof(float);
    float* scores  = (float*)w;                   w += B_ * P_ * sizeof(float);
    float* emb_s   = (float*)w;                   w += B_ * T_ * EMB_ * sizeof(float);
    float* att1    = (float*)w;                   w += (long long)B_ * P_ * ATT_ * sizeof(float);
    (void)ws_size; (void)n_in; (void)in_sizes; (void)out_size;

    auto gemm = [&](const float* A, int lda, const float* Bw, int ldb, const float* bias,
                    float* C, int ldc, int M, int N, int K,
                    const int* rowmap, int rpg, int accum,
                    const int* dl, int st) {
        const int tilesN = (N + 63) / 64;
        const int tilesM = (M + 15) / 16;
        dim3 grid(tilesN, (tilesM + 7) / 8);
        k_gemm_wmma<<<grid, 256, 0, stream>>>(A, lda, Bw, ldb, bias, C, ldc,
                                              M, N, K, rowmap, rpg, accum, dl, st);
    };

    // ---- prologue ----
    k_sort<<<1, 128, 0, stream>>>(cl, sind, dlen);
    k_addvec<<<8, 256, 0, stream>>>(l1_bih, l1_bhh, bias_l1, 2048);
    k_addvec<<<8, 256, 0, stream>>>(l2_bih, l2_bhh, bias_l2, 2048);
    k_addvec<<<2, 256, 0, stream>>>(aa_b, ad_b, bias_at, 512);
    k_zero<<<(B_ * DEC_ + 255) / 256, 256, 0, stream>>>(h2, B_ * DEC_);
    k_zero<<<(B_ * DEC_ + 255) / 256, 256, 0, stream>>>(c2, B_ * DEC_);
    k_embed<<<(B_ * T_ * EMB_ + 255) / 256, 256, 0, stream>>>(caps, sind, emb, emb_s);

    // x0 = attrs[sort] @ x0_w + x0_b
    gemm(attrs, DEC_, x0_w, EMB_, x0_b, x0, EMB_, B_, EMB_, DEC_, sind, 1, 0, nullptr, 0);
    // initial LSTM1: g1 = x0 @ l1_wih + (bih+bhh)   (h=0 -> skip whh)
    gemm(x0, EMB_, l1_wih, 4 * DEC_, bias_l1, g1, 4 * DEC_, B_, 4 * DEC_, EMB_,
         nullptr, 1, 0, nullptr, 0);
    k_lstm<<<(B_ * DEC_ + 255) / 256, 256, 0, stream>>>(g1, h1, c1, nullptr, 0, 0);
    // att1 = enc[sort] @ ae_w + ae_b  (fused gather via row remap, rpg = P)
    gemm(enc, ENC_, ae_w, ATT_, ae_b, att1, ATT_, B_ * P_, ATT_, ENC_,
         sind, P_, 0, nullptr, 0);

    // ---- 25 sequential decode steps ----
    for (int t = 0; t < T_; ++t) {
        const float* emb_t = emb_s + t * EMB_;   // row stride T_*EMB_

        // att23 = h1 @ aa_w + h2 @ ad_w + (aa_b + ad_b)
        gemm(h1, DEC_, aa_w, ATT_, bias_at, att23, ATT_, B_, ATT_, DEC_,
             nullptr, 1, 0, nullptr, 0);
        gemm(h2, DEC_, ad_w, ATT_, nullptr, att23, ATT_, B_, ATT_, DEC_,
             nullptr, 1, 1, nullptr, 0);

        k_scores<<<(B_ * P_ + 7) / 8, 256, 0, stream>>>(att1, att23, full_w, full_b,
                                                        scores, B_ * P_);
        k_softmax<<<B_, 256, 0, stream>>>(scores);
        k_awe<<<dim3(ENC_ / 256, B_), 256, 0, stream>>>(scores, enc, sind, awe);

        // LSTM1 gates: g1 = emb_t @ l1_wih + h1 @ l1_whh + bias
        gemm(emb_t, T_ * EMB_, l1_wih, 4 * DEC_, bias_l1, g1, 4 * DEC_,
             B_, 4 * DEC_, EMB_, nullptr, 1, 0, nullptr, 0);
        gemm(h1, DEC_, l1_whh, 4 * DEC_, nullptr, g1, 4 * DEC_,
             B_, 4 * DEC_, DEC_, nullptr, 1, 1, nullptr, 0);

        // LSTM2 gates: g2 = [emb_t, awe] @ l2_wih + h2 @ l2_whh + bias
        //   concat fused as two sub-K GEMMs into the same accumulator
        gemm(emb_t, T_ * EMB_, l2_wih, 4 * DEC_, bias_l2, g2, 4 * DEC_,
             B_, 4 * DEC_, EMB_, nullptr, 1, 0, nullptr, 0);
        gemm(awe, ENC_, l2_wih + (long long)EMB_ * 4 * DEC_, 4 * DEC_, nullptr, g2,
             4 * DEC_, B_, 4 * DEC_, ENC_, nullptr, 1, 1, nullptr, 0);
        gemm(h2, DEC_, l2_whh, 4 * DEC_, nullptr, g2, 4 * DEC_,
             B_, 4 * DEC_, DEC_, nullptr, 1, 1, nullptr, 0);

        // masked carry updates
        k_lstm<<<(B_ * DEC_ + 255) / 256, 256, 0, stream>>>(g1, h1, c1, dlen, t, 1);
        k_lstm<<<(B_ * DEC_ + 255) / 256, 256, 0, stream>>>(g2, h2, c2, dlen, t, 1);

        // preds[:, t, :] = (h2 @ fc_w + fc_b) * active   (fused mask + strided store)
        gemm(h2, DEC_, fc_w, V_, fc_b, out + (long long)t * V_, T_ * V_,
             B_, V_, DEC_, nullptr, 1, 0, dlen, t);
    }
}